// DySample_21483426414621
// MI455X (gfx1250) — compile-verified
//
#include <hip/hip_runtime.h>
#include <hip/hip_bf16.h>
#include <math.h>

// ---------------------------------------------------------------------------
// DySample (R=2, G=4) for MI455X / gfx1250.
//   x:      (8, 256, 128, 128) f32
//   W_off:  (8, 256), b_off: (8,)
//   W_wgt:  (4, 256), b_wgt: (4,)
//   out:    (8, 64, 256, 256) f32
// Kernel 1: 1x1-conv GEMM (M=131072, K=256, N=12 pad 16) via v_wmma_f32_16x16x4_f32
// Kernel 2: softmax + bilinear gather + weighted sum + fused pixel_shuffle,
//           LDS-transposed epilogue for full-cacheline output stores.
// ---------------------------------------------------------------------------

typedef __attribute__((ext_vector_type(2))) float v2f;
typedef __attribute__((ext_vector_type(8))) float v8f;

#define BATCH 8
#define CIN   256
#define HH    128
#define WW    128
#define HW    (HH * WW)          // 16384
#define NPAD  16                 // conv outputs padded 12 -> 16

// ---------------------------------------------------------------------------
// Kernel 1: per-pixel 1x1 conv, D = A(16xK) * B(Kx16) + bias, K=256.
//   A[m][k] = x[b][k][m]  (16 contiguous floats per k-row -> coalesced)
//   B[k][n] = Wcat[n][k], Wcat rows 0..7 = W_off, 8..11 = W_wgt, 12..15 = 0
// B staged in LDS as interleaved pairs: sBp[(k>>1)*32 + n*2 + (k&1)], so the
// per-step B fragment is ONE ds_load_b64 (ka is always even).
// One wave computes one 16-pixel tile; result stored pixel-major in ws.
// ---------------------------------------------------------------------------
__global__ void __launch_bounds__(256)
dysample_conv_wmma(const float* __restrict__ x,
                   const float* __restrict__ W_off,
                   const float* __restrict__ b_off,
                   const float* __restrict__ W_wgt,
                   const float* __restrict__ b_wgt,
                   float* __restrict__ ws) {
    __shared__ float sBp[CIN * NPAD];  // 16 KB, pair-interleaved layout

    const int t = threadIdx.x;
    // Stage B into LDS (4096 elements / 256 threads).
    {
        const int n  = t & 15;
        const int kb = t >> 4;               // 0..15
        #pragma unroll
        for (int i = 0; i < 16; ++i) {
            const int k = kb + (i << 4);     // covers k = 0..255
            float v = 0.0f;
            if (n < 8)       v = W_off[n * CIN + k];
            else if (n < 12) v = W_wgt[(n - 8) * CIN + k];
            sBp[(k >> 1) * 32 + n * 2 + (k & 1)] = v;
        }
    }
    __syncthreads();

    const int wave   = t >> 5;
    const int lane   = t & 31;
    const int laneLo = lane & 15;
    const int hi     = lane >> 4;            // 0: lanes 0-15, 1: lanes 16-31

    // 8192 tiles total (8 batches * 1024 tiles), 8 waves (tiles) per block.
    const int tile = blockIdx.x * 8 + wave;
    const int b    = tile >> 10;             // tile / 1024
    const int m0   = (tile & 1023) << 4;     // pixel base within batch

    const float* xb = x + (size_t)b * CIN * HW;

    v8f acc = {};
    // K-loop: 4 channels per WMMA step.
    // A-fragment layout (ISA 7.12.2, 32-bit A 16x4):
    //   VGPR0: lanes0-15 K=k0, lanes16-31 K=k0+2 ; VGPR1: K=k0+1 / K=k0+3
    for (int k0 = 0; k0 < CIN; k0 += 4) {
        const int ka = k0 + (hi << 1);       // even
        v2f a, bfrag;
        a.x = xb[(size_t)ka       * HW + m0 + laneLo];
        a.y = xb[(size_t)(ka + 1) * HW + m0 + laneLo];
        bfrag = *(const v2f*)&sBp[(ka >> 1) * 32 + laneLo * 2];  // one b64
        acc = __builtin_amdgcn_wmma_f32_16x16x4_f32(
            /*neg_a=*/false, a, /*neg_b=*/false, bfrag,
            /*c_mod=*/(short)0, acc, /*reuse_a=*/false, /*reuse_b=*/false);
    }

    // Epilogue: bias + pixel-major store.
    // C layout: VGPR v, lane L -> D[M = v + 8*(L>=16)][N = L&15]
    const int   n    = laneLo;
    const float bias = (n < 8) ? b_off[n] : ((n < 12) ? b_wgt[n - 8] : 0.0f);
    float* wsb = ws + ((size_t)b * HW + m0) * NPAD;
    #pragma unroll
    for (int v = 0; v < 8; ++v) {
        const int m = v + (hi << 3);
        wsb[(size_t)m * NPAD + n] = acc[v] + bias;
    }
}

// ---------------------------------------------------------------------------
// Kernel 2: one 1x16 low-res pixel strip per block (256 threads).
//  Phase 1 (threads 0..63): per (pixel,group) metadata -> 16 (idx, weight)
//           pairs per pixel in LDS (softmax_g * bilinear weight, OOB -> 0).
//  Phase 2 (thread c = channel): 16-term gather-FMA per pixel, result to LDS.
//  Phase 3: LDS transpose -> float4 stores; each output plane row segment is
//           32 contiguous floats (full 128B lines across lane pairs).
// ---------------------------------------------------------------------------
__global__ void __launch_bounds__(256)
dysample_sample(const float* __restrict__ x,
                const float* __restrict__ meta,
                float* __restrict__ out) {
    __shared__ int   s_idx[16 * 16];
    __shared__ float s_w[16 * 16];
    __shared__ float s_val[CIN * 17];        // stride 17: bank-conflict-free

    const int blk = blockIdx.x;              // 8192 blocks
    const int b   = blk >> 10;               // / 1024
    const int rem = blk & 1023;
    const int th  = rem >> 3;                // row 0..127
    const int tw  = (rem & 7) << 4;          // col base: 0,16,...,112

    const int t = threadIdx.x;

    // Prefetch this channel's gather neighborhood (global_prefetch_b8).
    {
        const float* xpf = x + ((size_t)b * CIN + t) * HW + th * WW + tw;
        __builtin_prefetch(xpf, 0, 1);
    }

    if (t < 64) {
        const int p = t >> 2;                // pixel in strip: 0..15
        const int g = t & 3;                 // group: 0..3
        const int ph = th;
        const int pw = tw + p;
        const float* mp = meta + ((size_t)b * HW + ph * WW + pw) * NPAD;

        // softmax over the 4 logits (channels 8..11)
        const float l0 = mp[8], l1 = mp[9], l2 = mp[10], l3 = mp[11];
        const float mx = fmaxf(fmaxf(l0, l1), fmaxf(l2, l3));
        const float e0 = expf(l0 - mx), e1 = expf(l1 - mx);
        const float e2 = expf(l2 - mx), e3 = expf(l3 - mx);
        const float inv = 1.0f / (e0 + e1 + e2 + e3);
        const float eg  = (g == 0) ? e0 : (g == 1) ? e1 : (g == 2) ? e2 : e3;
        const float wsm = eg * inv;

        // unnormalized position (align_corners=False): gx = (pw+ox)*W/(W-1) - 0.5
        const float S  = (float)WW / (float)(WW - 1);
        const float ox = mp[2 * g], oy = mp[2 * g + 1];
        const float gx = ((float)pw + ox) * S - 0.5f;
        const float gy = ((float)ph + oy) * S - 0.5f;

        const float x0f = floorf(gx), y0f = floorf(gy);
        const float wx1 = gx - x0f, wx0 = 1.0f - wx1;
        const float wy1 = gy - y0f, wy0 = 1.0f - wy1;
        const int x0 = (int)x0f, y0 = (int)y0f;
        const int x1 = x0 + 1,   y1 = y0 + 1;

        const int base = p * 16 + g * 4;
        const int   cx[4] = { x0, x1, x0, x1 };
        const int   cy[4] = { y0, y0, y1, y1 };
        const float cw[4] = { wx0 * wy0, wx1 * wy0, wx0 * wy1, wx1 * wy1 };
        #pragma unroll
        for (int j = 0; j < 4; ++j) {
            const int  xi = cx[j], yi = cy[j];
            const bool ok = (xi >= 0) & (xi < WW) & (yi >= 0) & (yi < HH);
            const int  xc = min(max(xi, 0), WW - 1);
            const int  yc = min(max(yi, 0), HH - 1);
            s_idx[base + j] = yc * WW + xc;
            s_w[base + j]   = ok ? (wsm * cw[j]) : 0.0f;
        }
    }
    __syncthreads();

    // Phase 2: gather + weighted sum per channel.
    {
        const int c = t;                     // channel 0..255
        const float* xb = x + ((size_t)b * CIN + c) * HW;
        #pragma unroll 4
        for (int p = 0; p < 16; ++p) {
            float accv = 0.0f;
            #pragma unroll
            for (int j = 0; j < 16; ++j)
                accv += s_w[p * 16 + j] * xb[s_idx[p * 16 + j]];
            s_val[c * 17 + p] = accv;
        }
    }
    __syncthreads();

    // Phase 3: pixel_shuffle transpose + vectorized stores.
    // Output region: 64 planes x 2 rows (2*th+r1) x 32 cols (2*tw..2*tw+31).
    // Thread t -> (co, r1, half-row of 16 floats), written as 4x float4.
    {
        const int rowid = t >> 1;            // 0..127
        const int half  = t & 1;             // 0: cols 0-15, 1: cols 16-31
        const int co    = rowid >> 1;        // 0..63
        const int r1    = rowid & 1;
        float* orow = out + (((size_t)b * 64 + co) * (2 * HH) + 2 * th + r1)
                          * (2 * WW) + 2 * tw + 16 * half;
        #pragma unroll
        for (int ccj = 0; ccj < 4; ++ccj) {
            float4 v4;
            #pragma unroll
            for (int e = 0; e < 4; ++e) {
                const int col = 16 * half + ccj * 4 + e; // 0..31
                const int p   = col >> 1;
                const int r2  = col & 1;
                const int c   = co * 4 + r1 * 2 + r2;
                ((float*)&v4)[e] = s_val[c * 17 + p];
            }
            *(float4*)(orow + ccj * 4) = v4;
        }
    }
}

// ---------------------------------------------------------------------------
extern "C" void kernel_launch(void* const* d_in, const int* in_sizes, int n_in,
                              void* d_out, int out_size, void* d_ws, size_t ws_size,
                              hipStream_t stream) {
    (void)in_sizes; (void)n_in; (void)out_size; (void)ws_size;
    const float* x     = (const float*)d_in[0];
    const float* W_off = (const float*)d_in[1];
    const float* b_off = (const float*)d_in[2];
    const float* W_wgt = (const float*)d_in[3];
    const float* b_wgt = (const float*)d_in[4];
    float* out = (float*)d_out;
    float* ws  = (float*)d_ws;               // needs 8*16384*16*4 = 8 MB

    // Kernel 1: 8192 16-pixel tiles, 8 waves (tiles) per 256-thread block.
    dysample_conv_wmma<<<1024, 256, 0, stream>>>(x, W_off, b_off, W_wgt, b_wgt, ws);
    // Kernel 2: one 1x16 pixel strip per block.
    dysample_sample<<<8192, 256, 0, stream>>>(x, ws, out);
}